// GCN_46626164965923
// MI455X (gfx1250) — compile-verified
//
#include <hip/hip_runtime.h>
#include <math.h>

typedef __attribute__((ext_vector_type(2))) float v2f;
typedef __attribute__((ext_vector_type(8))) float v8f;

// ---------------------------------------------------------------------------
// Fills (float4-wide)
// ---------------------------------------------------------------------------
__global__ void fill4_kernel(float4* __restrict__ p, float v, size_t n4) {
    size_t i = (size_t)blockIdx.x * blockDim.x + threadIdx.x;
    if (i < n4) p[i] = make_float4(v, v, v, v);
}

__global__ void fill_f32_kernel(float* __restrict__ p, float v, size_t n) {
    size_t i = (size_t)blockIdx.x * blockDim.x + threadIdx.x;
    if (i < n) p[i] = v;
}

__global__ void deg_kernel(const int* __restrict__ dst, float* __restrict__ deg, int E) {
    int e = blockIdx.x * blockDim.x + threadIdx.x;
    if (e < E) unsafeAtomicAdd(&deg[dst[e]], 1.0f);
}

__global__ void rsqrt_kernel(float* __restrict__ deg, int n) {
    int i = blockIdx.x * blockDim.x + threadIdx.x;
    if (i < n) deg[i] = rsqrtf(deg[i]);   // deg >= 1 always (self-loop)
}

// ---------------------------------------------------------------------------
// Pack W[K,256] into row-pair interleaved form:
//   Wp[(k>>1)*256 + n] = { W[k,n], W[k+1,n] }   (k even)
// so a WMMA B fragment is one contiguous float2 (global_load_b64).
// ---------------------------------------------------------------------------
__global__ void pack_w_kernel(const float* __restrict__ W, float2* __restrict__ Wp,
                              int K) {
    int idx = blockIdx.x * blockDim.x + threadIdx.x;    // over (K/2)*256
    int total = (K >> 1) * 256;
    if (idx < total) {
        int kp = idx >> 8;          // k/2
        int n  = idx & 255;
        float2 v;
        v.x = W[(size_t)(2 * kp + 0) * 256 + n];
        v.y = W[(size_t)(2 * kp + 1) * 256 + n];
        Wp[idx] = v;
    }
}

// ---------------------------------------------------------------------------
// fp32 WMMA GEMM: C[M,256] = X[M,K] @ W[K,256]  (W pre-packed as above)
// One wave computes a 16x256 output stripe (16 N-tiles) with
// v_wmma_f32_16x16x4_f32; A fragment reused across all 16 N-tiles.
// A and B fragments are each a single b64 load.
// ---------------------------------------------------------------------------
__global__ __launch_bounds__(128)
void gemm_wmma_f32(const float* __restrict__ X, const float2* __restrict__ Wp,
                   float* __restrict__ C, int M, int K) {
    const int lane  = threadIdx.x & 31;
    const int wave  = threadIdx.x >> 5;
    const int mtile = blockIdx.x * 4 + wave;        // wave-uniform
    if (mtile * 16 >= M) return;                     // whole wave exits -> EXEC all-1s at WMMA

    const int mrow = lane & 15;                      // A fragment row (M within tile)
    const int hi   = lane >> 4;                      // half-wave selector

    // A: lanes 0-15 hold K={k0,k0+1}, lanes 16-31 hold K={k0+2,k0+3}
    const float2* xrow = (const float2*)(X + (size_t)(mtile * 16 + mrow) * K) + hi;

    v8f acc[16];
#pragma unroll
    for (int t = 0; t < 16; ++t) acc[t] = (v8f)0.0f;

    for (int k0 = 0; k0 < K; k0 += 4) {
        float2 av = xrow[k0 >> 1];
        v2f a; a.x = av.x; a.y = av.y;
        // B fragment: rows {k0+2*hi, k0+2*hi+1} at column n = t*16 + lane&15
        const float2* wp = Wp + (size_t)((k0 >> 1) + hi) * 256 + (lane & 15);
#pragma unroll
        for (int t = 0; t < 16; ++t) {
            float2 bv = wp[t * 16];
            v2f b; b.x = bv.x; b.y = bv.y;
            acc[t] = __builtin_amdgcn_wmma_f32_16x16x4_f32(
                false, a, false, b, (short)0, acc[t], false, false);
        }
    }

    // C/D layout: VGPR r holds M = r (lanes 0-15) / M = 8+r (lanes 16-31), N = lane&15
    const int mbase = mtile * 16 + 8 * hi;
#pragma unroll
    for (int t = 0; t < 16; ++t) {
        float* crow = C + (size_t)mbase * 256 + t * 16 + (lane & 15);
#pragma unroll
        for (int r = 0; r < 8; ++r) crow[(size_t)r * 256] = acc[t][r];
    }
}

// ---------------------------------------------------------------------------
// Edge scatter: Out[dst] += H[src] * dis[src]*dis[dst]; self-loops appended.
// One wave per edge, float4 loads (b128) + hardware f32 atomics.
// ---------------------------------------------------------------------------
__global__ __launch_bounds__(256)
void scatter_kernel(const float* __restrict__ H, float* __restrict__ Out,
                    const int* __restrict__ src, const int* __restrict__ dst,
                    const float* __restrict__ dis, int E, int Nn) {
    const int widx = blockIdx.x * 8 + (threadIdx.x >> 5);
    const int lane = threadIdx.x & 31;
    const int total = E + Nn;
    if (widx >= total) return;
    int s, d;
    if (widx < E) { s = src[widx]; d = dst[widx]; }
    else          { s = widx - E; d = s; }
    const float nrm = dis[s] * dis[d];
    const float4* hrow = (const float4*)(H + (size_t)s * 256);
    float* orow = Out + (size_t)d * 256;
#pragma unroll
    for (int it = 0; it < 2; ++it) {
        int c = it * 32 + lane;             // float4 index, 64 per row
        float4 hv = hrow[c];
        unsafeAtomicAdd(&orow[4 * c + 0], hv.x * nrm);
        unsafeAtomicAdd(&orow[4 * c + 1], hv.y * nrm);
        unsafeAtomicAdd(&orow[4 * c + 2], hv.z * nrm);
        unsafeAtomicAdd(&orow[4 * c + 3], hv.w * nrm);
    }
}

// ---------------------------------------------------------------------------
// H = relu(H + b)   (feature width 256, float4-wide)
// ---------------------------------------------------------------------------
__global__ void bias_relu_kernel(float4* __restrict__ H, const float4* __restrict__ b,
                                 size_t n4) {
    size_t i = (size_t)blockIdx.x * blockDim.x + threadIdx.x;
    if (i < n4) {
        float4 v = H[i];
        float4 bb = b[i & 63];              // 64 float4 = 256 floats
        v.x = fmaxf(v.x + bb.x, 0.0f);
        v.y = fmaxf(v.y + bb.y, 0.0f);
        v.z = fmaxf(v.z + bb.z, 0.0f);
        v.w = fmaxf(v.w + bb.w, 0.0f);
        H[i] = v;
    }
}

// ---------------------------------------------------------------------------
// Global mean-pool accumulation: sums[batch[n]] += H[n]; cnts[batch[n]] += 1
// ---------------------------------------------------------------------------
__global__ __launch_bounds__(256)
void pool_kernel(const float* __restrict__ H, const int* __restrict__ batch,
                 float* __restrict__ sums, float* __restrict__ cnts, int Nn) {
    int node = blockIdx.x;
    if (node >= Nn) return;
    int g = batch[node];
    int c = threadIdx.x;
    unsafeAtomicAdd(&sums[g * 256 + c], H[(size_t)node * 256 + c]);
    if (c == 0) unsafeAtomicAdd(&cnts[g], 1.0f);
}

// ---------------------------------------------------------------------------
// Head: pooled = sums/max(cnt,1); logits = pooled@Wc + bc; log_softmax
// ---------------------------------------------------------------------------
__global__ __launch_bounds__(64)
void head_kernel(const float* __restrict__ sums, const float* __restrict__ cnts,
                 const float* __restrict__ Wc, const float* __restrict__ bc,
                 float* __restrict__ out, int nGraphs) {
    int g = threadIdx.x;
    if (g >= nGraphs) return;
    float inv = 1.0f / fmaxf(cnts[g], 1.0f);
    float logits[10];
#pragma unroll
    for (int j = 0; j < 10; ++j) logits[j] = bc[j];
    for (int c = 0; c < 256; ++c) {
        float p = sums[g * 256 + c] * inv;
#pragma unroll
        for (int j = 0; j < 10; ++j) logits[j] += p * Wc[c * 10 + j];
    }
    float mx = logits[0];
#pragma unroll
    for (int j = 1; j < 10; ++j) mx = fmaxf(mx, logits[j]);
    float se = 0.0f;
#pragma unroll
    for (int j = 0; j < 10; ++j) se += expf(logits[j] - mx);
    float lse = mx + logf(se);
#pragma unroll
    for (int j = 0; j < 10; ++j) out[g * 10 + j] = logits[j] - lse;
}

// ---------------------------------------------------------------------------
extern "C" void kernel_launch(void* const* d_in, const int* in_sizes, int n_in,
                              void* d_out, int out_size, void* d_ws, size_t ws_size,
                              hipStream_t stream) {
    const float* x     = (const float*)d_in[0];
    const int*   ei    = (const int*)  d_in[1];
    const int*   batch = (const int*)  d_in[2];
    const float* W1 = (const float*)d_in[3];
    const float* b1 = (const float*)d_in[4];
    const float* W2 = (const float*)d_in[5];
    const float* b2 = (const float*)d_in[6];
    const float* W3 = (const float*)d_in[7];
    const float* b3 = (const float*)d_in[8];
    const float* Wc = (const float*)d_in[9];
    const float* bc = (const float*)d_in[10];
    float* out = (float*)d_out;

    const int Nn = in_sizes[0] / 128;   // 100000 nodes
    const int E  = in_sizes[1] / 2;     // 1600000 edges
    const int nGraphs = 64;

    const int* srcIdx = ei;
    const int* dstIdx = ei + E;

    // ---- workspace carve-out ----
    char* wsb = (char*)d_ws;
    size_t off = 0;
    auto carve = [&](size_t bytes) -> void* {
        off = (off + 255) & ~(size_t)255;
        void* p = (void*)(wsb + off);
        off += bytes;
        return p;
    };
    float*  dis  = (float*) carve((size_t)Nn * sizeof(float));        // deg -> dis (in place)
    float*  bufA = (float*) carve((size_t)Nn * 256 * sizeof(float));  // GEMM output
    float*  bufB = (float*) carve((size_t)Nn * 256 * sizeof(float));  // scatter target
    float*  sums = (float*) carve((size_t)nGraphs * 256 * sizeof(float));
    float*  cnts = (float*) carve((size_t)nGraphs * sizeof(float));
    float2* Wp1  = (float2*)carve((size_t)(128 / 2) * 256 * sizeof(float2));
    float2* Wp2  = (float2*)carve((size_t)(256 / 2) * 256 * sizeof(float2));
    float2* Wp3  = (float2*)carve((size_t)(256 / 2) * 256 * sizeof(float2));
    (void)ws_size;

    const size_t feats  = (size_t)Nn * 256;
    const size_t feats4 = feats / 4;
    const int THR = 256;
    const int gridNodes  = (Nn + THR - 1) / THR;
    const int gridEdges  = (E + THR - 1) / THR;
    const int gridFeats4 = (int)((feats4 + THR - 1) / THR);
    const int mtiles     = (Nn + 15) / 16;            // 6250
    const int gemmGrid   = (mtiles + 3) / 4;          // 4 waves/block, 1 mtile/wave
    const int scatGrid   = (E + Nn + 7) / 8;          // 8 waves/block, 1 edge/wave

    // ---- weight pre-pack (tiny, once per call) ----
    pack_w_kernel<<<(64 * 256 + THR - 1) / THR, THR, 0, stream>>>(W1, Wp1, 128);
    pack_w_kernel<<<(128 * 256 + THR - 1) / THR, THR, 0, stream>>>(W2, Wp2, 256);
    pack_w_kernel<<<(128 * 256 + THR - 1) / THR, THR, 0, stream>>>(W3, Wp3, 256);

    // ---- gcn_norm: deg (with self-loop) -> dis = deg^-1/2 ----
    fill_f32_kernel<<<gridNodes, THR, 0, stream>>>(dis, 1.0f, (size_t)Nn);
    deg_kernel<<<gridEdges, THR, 0, stream>>>(dstIdx, dis, E);
    rsqrt_kernel<<<gridNodes, THR, 0, stream>>>(dis, Nn);

    // ---- layer 1: h = relu(scatter(x@W1) + b1) ----
    gemm_wmma_f32<<<gemmGrid, 128, 0, stream>>>(x, Wp1, bufA, Nn, 128);
    fill4_kernel<<<gridFeats4, THR, 0, stream>>>((float4*)bufB, 0.0f, feats4);
    scatter_kernel<<<scatGrid, THR, 0, stream>>>(bufA, bufB, srcIdx, dstIdx, dis, E, Nn);
    bias_relu_kernel<<<gridFeats4, THR, 0, stream>>>((float4*)bufB, (const float4*)b1, feats4);

    // ---- layer 2 ----
    gemm_wmma_f32<<<gemmGrid, 128, 0, stream>>>(bufB, Wp2, bufA, Nn, 256);
    fill4_kernel<<<gridFeats4, THR, 0, stream>>>((float4*)bufB, 0.0f, feats4);
    scatter_kernel<<<scatGrid, THR, 0, stream>>>(bufA, bufB, srcIdx, dstIdx, dis, E, Nn);
    bias_relu_kernel<<<gridFeats4, THR, 0, stream>>>((float4*)bufB, (const float4*)b2, feats4);

    // ---- layer 3 ----
    gemm_wmma_f32<<<gemmGrid, 128, 0, stream>>>(bufB, Wp3, bufA, Nn, 256);
    fill4_kernel<<<gridFeats4, THR, 0, stream>>>((float4*)bufB, 0.0f, feats4);
    scatter_kernel<<<scatGrid, THR, 0, stream>>>(bufA, bufB, srcIdx, dstIdx, dis, E, Nn);
    bias_relu_kernel<<<gridFeats4, THR, 0, stream>>>((float4*)bufB, (const float4*)b3, feats4);

    // ---- mean pool + classifier head ----
    fill4_kernel<<<(nGraphs * 64 + THR - 1) / THR, THR, 0, stream>>>(
        (float4*)sums, 0.0f, (size_t)nGraphs * 64);
    fill_f32_kernel<<<1, 64, 0, stream>>>(cnts, 0.0f, (size_t)nGraphs);
    pool_kernel<<<Nn, 256, 0, stream>>>(bufB, batch, sums, cnts, Nn);
    head_kernel<<<1, 64, 0, stream>>>(sums, cnts, Wc, bc, out, nGraphs);
}